// GMRFSamplingLoss_15006615732608
// MI455X (gfx1250) — compile-verified
//
#include <hip/hip_runtime.h>
#include <hip/hip_bf16.h>
#include <math.h>

typedef float v2f __attribute__((ext_vector_type(2)));
typedef float v8f __attribute__((ext_vector_type(8)));

#define D_MEAN 32
#define RANK 16

// ---------------------------------------------------------------- zero
__global__ void k_zero(float* __restrict__ p, int n) {
  int i = blockIdx.x * blockDim.x + threadIdx.x;
  int stride = gridDim.x * blockDim.x;
  for (; i < n; i += stride) p[i] = 0.0f;
}

// ---------------------------------------------------------------- degree
__global__ void k_degree(const long long* __restrict__ row,
                         float* __restrict__ deg, int E) {
  int i = blockIdx.x * blockDim.x + threadIdx.x;
  int stride = gridDim.x * blockDim.x;
  for (; i < E; i += stride) {
    atomicAdd(&deg[(int)row[i]], 1.0f);
  }
}

// ---------------------------------------------------------------- dinv + sum(z_mean^2)
__global__ void k_node(const float* __restrict__ zmean,
                       const float* __restrict__ deg,
                       float* __restrict__ dinv,
                       float* __restrict__ sum_mean2, int N) {
  int i = blockIdx.x * blockDim.x + threadIdx.x;
  int stride = gridDim.x * blockDim.x;
  float s = 0.0f;
  for (; i < N; i += stride) {
    float d = deg[i];
    dinv[i] = (d > 0.0f) ? rsqrtf(d) : 0.0f;
    const float4* r = (const float4*)(zmean + (size_t)i * D_MEAN);
#pragma unroll
    for (int j = 0; j < D_MEAN / 4; ++j) {
      float4 v = r[j];
      s += v.x * v.x + v.y * v.y + v.z * v.z + v.w * v.w;
    }
  }
  for (int off = 16; off > 0; off >>= 1) s += __shfl_down(s, off, 32);
  __shared__ float bs;
  if (threadIdx.x == 0) bs = 0.0f;
  __syncthreads();
  if ((threadIdx.x & 31) == 0) atomicAdd(&bs, s);
  __syncthreads();
  if (threadIdx.x == 0) atomicAdd(sum_mean2, bs);
}

// ---------------------------------------------------------------- edge pass (L2-gather bound)
__global__ void k_edges(const float* __restrict__ zmean,
                        const float* __restrict__ zstd,
                        const long long* __restrict__ ei,
                        const float* __restrict__ dinv,
                        float* __restrict__ scal, int E) {
  int i = blockIdx.x * blockDim.x + threadIdx.x;
  int stride = gridDim.x * blockDim.x;
  float am = 0.0f, as = 0.0f, aw = 0.0f;
  for (; i < E; i += stride) {
    long long r = ei[i];
    long long c = ei[(size_t)E + i];
    float w = dinv[r] * dinv[c];
    const float4* mr = (const float4*)(zmean + (size_t)r * D_MEAN);
    const float4* mc = (const float4*)(zmean + (size_t)c * D_MEAN);
    float dm = 0.0f;
#pragma unroll
    for (int j = 0; j < D_MEAN / 4; ++j) {
      float4 a = mr[j], b = mc[j];
      dm += a.x * b.x + a.y * b.y + a.z * b.z + a.w * b.w;
    }
    const float4* sr = (const float4*)(zstd + (size_t)r * RANK);
    const float4* sc = (const float4*)(zstd + (size_t)c * RANK);
    float dsv = 0.0f;
#pragma unroll
    for (int j = 0; j < RANK / 4; ++j) {
      float4 a = sr[j], b = sc[j];
      dsv += a.x * b.x + a.y * b.y + a.z * b.z + a.w * b.w;
    }
    am += w * dm;
    as += w * dsv;
    if (r == c) aw += w;
  }
  for (int off = 16; off > 0; off >>= 1) {
    am += __shfl_down(am, off, 32);
    as += __shfl_down(as, off, 32);
    aw += __shfl_down(aw, off, 32);
  }
  __shared__ float b3[3];
  if (threadIdx.x < 3) b3[threadIdx.x] = 0.0f;
  __syncthreads();
  if ((threadIdx.x & 31) == 0) {
    atomicAdd(&b3[0], am);
    atomicAdd(&b3[1], as);
    atomicAdd(&b3[2], aw);
  }
  __syncthreads();
  if (threadIdx.x < 3) atomicAdd(&scal[1 + threadIdx.x], b3[threadIdx.x]);
}

// ---------------------------------------------------------------- Gram G = S^T S via WMMA f32 16x16x4
// Each wave accumulates a private 16x16 f32 tile (8 VGPRs) over strided
// 16-row super-chunks (4 WMMAs each), then LDS-reduces and one global
// atomicAdd per element per block.
__global__ void k_gram(const float* __restrict__ S, float* __restrict__ G, int nsup) {
  const int lane = threadIdx.x & 31;
  const int waveInBlock = threadIdx.x >> 5;
  const int wavesPerBlock = blockDim.x >> 5;
  const int gwave = blockIdx.x * wavesPerBlock + waveInBlock;
  const int nwaves = gridDim.x * wavesPerBlock;
  const int halfsel = lane >> 4;  // 0 = lanes 0-15, 1 = lanes 16-31
  const int m = lane & 15;

  v8f acc = {};
  for (int s = gwave; s < nsup; s += nwaves) {
    int k0 = s * 16;
#pragma unroll
    for (int q = 0; q < 4; ++q) {
      int kk = k0 + q * 4;
#if __has_builtin(__builtin_amdgcn_wmma_f32_16x16x4_f32)
      // A (16x4, MxK): lane<16 -> K=kk,kk+1 ; lane>=16 -> K=kk+2,kk+3 ; M = lane&15
      v2f a;
      a.x = S[(size_t)(kk + 2 * halfsel) * RANK + m];
      a.y = S[(size_t)(kk + 2 * halfsel + 1) * RANK + m];
      // B (4x16, KxN): vgpr0: lanes0-15 K=kk, lanes16-31 K=kk+1;
      //                vgpr1: lanes0-15 K=kk+2, lanes16-31 K=kk+3; N = lane&15
      v2f b;
      b.x = S[(size_t)(kk + halfsel) * RANK + m];
      b.y = S[(size_t)(kk + 2 + halfsel) * RANK + m];
      acc = __builtin_amdgcn_wmma_f32_16x16x4_f32(
          /*neg_a=*/false, a, /*neg_b=*/false, b,
          /*c_mod=*/(short)0, acc, /*reuse_a=*/false, /*reuse_b=*/false);
#else
      // VALU fallback matching the C/D layout: lane owns (mm = i + 8*halfsel, n = lane&15)
#pragma unroll
      for (int k = 0; k < 4; ++k) {
#pragma unroll
        for (int i = 0; i < 8; ++i) {
          int mm = i + 8 * halfsel;
          acc[i] += S[(size_t)(kk + k) * RANK + mm] * S[(size_t)(kk + k) * RANK + m];
        }
      }
#endif
    }
  }

  __shared__ float gs[256];
  for (int i = threadIdx.x; i < 256; i += blockDim.x) gs[i] = 0.0f;
  __syncthreads();
  // C/D layout: VGPR i, lanes 0-15 -> M=i, lanes 16-31 -> M=i+8; N = lane&15
#pragma unroll
  for (int i = 0; i < 8; ++i) {
    int mm = i + 8 * halfsel;
    atomicAdd(&gs[mm * 16 + m], acc[i]);
  }
  __syncthreads();
  for (int i = threadIdx.x; i < 256; i += blockDim.x) atomicAdd(&G[i], gs[i]);
}

// ---------------------------------------------------------------- finalize (1 thread)
__global__ void k_finalize(const float* __restrict__ zstd,
                           const float* __restrict__ G,
                           const float* __restrict__ scal,
                           float* __restrict__ out, int N, int rem_start) {
  if (threadIdx.x != 0 || blockIdx.x != 0) return;
  float M[RANK][RANK];
  for (int i = 0; i < RANK; ++i)
    for (int j = 0; j < RANK; ++j) M[i][j] = G[i * RANK + j];
  // remainder rows not covered by 16-row super-chunks
  for (int r = rem_start; r < N; ++r) {
    for (int i = 0; i < RANK; ++i) {
      float si = zstd[(size_t)r * RANK + i];
      for (int j = 0; j < RANK; ++j) M[i][j] += si * zstd[(size_t)r * RANK + j];
    }
  }
  float trG = 0.0f;
  for (int i = 0; i < RANK; ++i) trG += M[i][i];  // == sum(z_std^2)

  // l1 = slogdet(I + M/beta), beta = 1 -> Cholesky of SPD 16x16
  float L[RANK][RANK];
  float l1 = 0.0f;
  for (int j = 0; j < RANK; ++j) {
    float d = 1.0f + M[j][j];
    for (int k = 0; k < j; ++k) d -= L[j][k] * L[j][k];
    float Ljj = sqrtf(d);
    L[j][j] = Ljj;
    l1 += 2.0f * logf(Ljj);
    float inv = 1.0f / Ljj;
    for (int i = j + 1; i < RANK; ++i) {
      float v = M[i][j];
      for (int k = 0; k < j; ++k) v -= L[i][k] * L[j][k];
      L[i][j] = v * inv;
    }
  }

  float sum_mean2 = scal[0];
  float wdot_mean = scal[1];
  float wdot_std  = scal[2];
  float self_w    = scal[3];

  float trace_L = (float)N - self_w;
  float l2 = (trG - wdot_std) + 1.0f * trace_L;  // BETA = 1
  float l3 = sum_mean2 - wdot_mean;
  out[0] = (l3 / (float)D_MEAN + l2 - l1) / (2.0f * (float)N);
}

// ---------------------------------------------------------------- launch
extern "C" void kernel_launch(void* const* d_in, const int* in_sizes, int n_in,
                              void* d_out, int out_size, void* d_ws, size_t ws_size,
                              hipStream_t stream) {
  const float* zmean = (const float*)d_in[0];
  const float* zstd  = (const float*)d_in[1];
  const long long* ei = (const long long*)d_in[2];  // int64 edge_index [2, E]

  const int N = in_sizes[0] / D_MEAN;
  const int E = in_sizes[2] / 2;

  float* ws   = (float*)d_ws;
  float* deg  = ws;                 // [N]
  float* dinv = ws + N;             // [N]
  float* G    = ws + 2 * (size_t)N; // [256]
  float* scal = G + 256;            // [4]: sum_mean2, wdot_mean, wdot_std, self_w

  // zero accumulators (deg + G + scalars)
  k_zero<<<dim3((N + 255) / 256), dim3(256), 0, stream>>>(deg, N);
  k_zero<<<dim3(2), dim3(256), 0, stream>>>(G, 260);

  // degree counts from row index
  k_degree<<<dim3(1024), dim3(256), 0, stream>>>(ei, deg, E);

  // dinv + sum(z_mean^2)
  k_node<<<dim3(512), dim3(256), 0, stream>>>(zmean, deg, dinv, scal, N);

  // edge-wise weighted dot products + self-loop weight
  k_edges<<<dim3(2048), dim3(256), 0, stream>>>(zmean, zstd, ei, dinv, scal, E);

  // Gram matrix z_std^T z_std via WMMA (16-row super-chunks)
  const int nsup = N >> 4;
  k_gram<<<dim3(64), dim3(256), 0, stream>>>(zstd, G, nsup);

  // Cholesky logdet + final scalar assembly
  k_finalize<<<dim3(1), dim3(1), 0, stream>>>(zstd, G, scal, (float*)d_out, N, nsup * 16);
}